// AttenHead_1580547972770
// MI455X (gfx1250) — compile-verified
//
#include <hip/hip_runtime.h>
#include <math.h>

typedef __attribute__((ext_vector_type(2))) float v2f;
typedef __attribute__((ext_vector_type(8))) float v8f;

#define FDIM 512
#define NH   8
#define FATT 64
#define NX   2048
#define NP   4096
#define NTOT (NX + NP)

// ---------------------------------------------------------------------------
// WMMA f32 16x16x4 step: D = A(16x4) * B(4x16) + C  (fp32 everywhere)
// Per-lane fragment layout (wave32):
//   A: lane<16 -> {A[r][k+0], A[r][k+1]} ; lane>=16 -> {A[r][k+2], A[r][k+3]}
//   B: same pattern with rows = output column n (we feed B row-major as NxK)
//   C/D: vgpr j holds C[j + 8*(lane>>4)][lane&15]
// ---------------------------------------------------------------------------
__device__ __forceinline__ v8f wmma4(v2f a, v2f b, v8f c) {
  return __builtin_amdgcn_wmma_f32_16x16x4_f32(
      /*neg_a=*/false, a, /*neg_b=*/false, b,
      /*c_mod=*/(short)0, c, /*reuse_a=*/false, /*reuse_b=*/false);
}

// Computes a 16 x (16*NSUB) tile of C = A * B^T.
// A: M x K row-major (row stride sA elems), rows m0..m0+15
// B: N x K row-major (row stride sB elems), rows n0..n0+16*NSUB-1
template <int NSUB>
__device__ __forceinline__ void gemm_tile(const float* __restrict__ A, int sA, int m0,
                                          const float* __restrict__ B, int sB, int n0,
                                          int K, v8f* acc) {
  const int lane = threadIdx.x & 31;
  const int half = lane >> 4;   // 0 or 1
  const int r    = lane & 15;
  const float* ar  = A + (size_t)(m0 + r) * (size_t)sA + 2 * half;
  const float* br0 = B + (size_t)(n0 + r) * (size_t)sB + 2 * half;
  for (int k = 0; k < K; k += 4) {
    v2f a;
    a.x = ar[k];
    a.y = ar[k + 1];
#pragma unroll
    for (int s = 0; s < NSUB; ++s) {
      const float* br = br0 + (size_t)(s * 16) * (size_t)sB;
      v2f b;
      b.x = br[k];
      b.y = br[k + 1];
      acc[s] = wmma4(a, b, acc[s]);
    }
  }
}

// ---------------------------------------------------------------------------
// Kernel 1: femb = [fx_in ; fp0] @ Wembd^T + bembd
// Writes fxE[h][x][e], fpE[h][p][e], fpT[h][e][p].
// grid = (FDIM/64, NTOT/16), block = 32 (one wave)
// ---------------------------------------------------------------------------
__global__ void k_embed(const float* __restrict__ fx_in, const float* __restrict__ fp0,
                        const float* __restrict__ Wembd, const float* __restrict__ bembd,
                        float* __restrict__ fxE, float* __restrict__ fpE,
                        float* __restrict__ fpT) {
  const int n0 = blockIdx.x * 64;
  const int m0 = blockIdx.y * 16;
  const bool isX = (m0 < NX);
  const float* A = isX ? fx_in : fp0;
  const int mbase = isX ? m0 : (m0 - NX);

  v8f acc[4] = {};
  gemm_tile<4>(A, FDIM, mbase, Wembd, FDIM, n0, FDIM, acc);

  const int lane = threadIdx.x & 31;
  const int half = lane >> 4;
  const int r    = lane & 15;
#pragma unroll
  for (int s = 0; s < 4; ++s) {
    const int n = n0 + s * 16 + r;
    const int h = n >> 6;
    const int e = n & 63;
    const float bb = bembd[n];
#pragma unroll
    for (int j = 0; j < 8; ++j) {
      const int m = mbase + j + 8 * half;
      const float v = acc[s][j] + bb;
      if (isX) {
        fxE[((size_t)h * NX + m) * FATT + e] = v;
      } else {
        fpE[((size_t)h * NP + m) * FATT + e] = v;
        fpT[((size_t)h * FATT + e) * NP + m] = v;
      }
    }
  }
}

// ---------------------------------------------------------------------------
// Kernel 2: scores = fxE[h] @ fpE[h]^T * 0.125, stored in w_out layout (x,h,p)
// grid = (NP/64, NX/16, NH), block = 32
// ---------------------------------------------------------------------------
__global__ void k_scores(const float* __restrict__ fxE, const float* __restrict__ fpE,
                         float* __restrict__ w) {
  const int h  = blockIdx.z;
  const int n0 = blockIdx.x * 64;  // p
  const int m0 = blockIdx.y * 16;  // x

  v8f acc[4] = {};
  gemm_tile<4>(fxE + (size_t)h * NX * FATT, FATT, m0,
               fpE + (size_t)h * NP * FATT, FATT, n0, FATT, acc);

  const int lane = threadIdx.x & 31;
  const int half = lane >> 4;
  const int r    = lane & 15;
#pragma unroll
  for (int s = 0; s < 4; ++s) {
    const int p = n0 + s * 16 + r;
#pragma unroll
    for (int j = 0; j < 8; ++j) {
      const int x = m0 + j + 8 * half;
      w[((size_t)x * NH + h) * NP + p] = acc[s][j] * 0.125f;
    }
  }
}

// ---------------------------------------------------------------------------
// Kernel 3: in-place softmax along the contiguous 4096-float rows of w
// grid = NX*NH blocks, block = 256
// ---------------------------------------------------------------------------
__global__ void k_softmax(float* __restrict__ w) {
  float* row = w + (size_t)blockIdx.x * NP;
  __shared__ float sred[256];
  const int tid = threadIdx.x;

  float m = -1e30f;
  for (int i = tid; i < NP; i += 256) m = fmaxf(m, row[i]);
  sred[tid] = m;
  __syncthreads();
  for (int s = 128; s > 0; s >>= 1) {
    if (tid < s) sred[tid] = fmaxf(sred[tid], sred[tid + s]);
    __syncthreads();
  }
  const float rmax = sred[0];
  __syncthreads();

  float sum = 0.f;
  for (int i = tid; i < NP; i += 256) {
    const float e = expf(row[i] - rmax);
    row[i] = e;
    sum += e;
  }
  sred[tid] = sum;
  __syncthreads();
  for (int s = 128; s > 0; s >>= 1) {
    if (tid < s) sred[tid] += sred[tid + s];
    __syncthreads();
  }
  const float inv = 1.0f / sred[0];
  for (int i = tid; i < NP; i += 256) row[i] *= inv;
}

// ---------------------------------------------------------------------------
// Kernel 4: fa_att = w[h] @ fp[h]  (K = 4096), plus copy fx into facat[:,64:128]
// facat[h][x][0:64] = fa_att, facat[h][x][64:128] = fxE
// grid = (NX/16, NH), block = 32
// ---------------------------------------------------------------------------
__global__ void k_attv(const float* __restrict__ w, const float* __restrict__ fpT,
                       const float* __restrict__ fxE, float* __restrict__ facat) {
  const int h  = blockIdx.y;
  const int m0 = blockIdx.x * 16;

  v8f acc[4] = {};
  // A = w rows (x), stride NH*NP, offset h*NP ; B = fpT[h] (64 x 4096)
  gemm_tile<4>(w + (size_t)h * NP, NH * NP, m0,
               fpT + (size_t)h * FATT * NP, NP, 0, NP, acc);

  const int lane = threadIdx.x & 31;
  const int half = lane >> 4;
  const int r    = lane & 15;
#pragma unroll
  for (int s = 0; s < 4; ++s) {
    const int e = s * 16 + r;
#pragma unroll
    for (int j = 0; j < 8; ++j) {
      const int x = m0 + j + 8 * half;
      facat[((size_t)h * NX + x) * 128 + e] = acc[s][j];
    }
  }
  // copy fx half of the concat for this 16-row stripe
  for (int idx = lane; idx < 16 * FATT; idx += 32) {
    const int x = m0 + (idx >> 6);
    const int e = idx & 63;
    facat[((size_t)h * NX + x) * 128 + 64 + e] =
        fxE[((size_t)h * NX + x) * FATT + e];
  }
}

// ---------------------------------------------------------------------------
// Kernel 5: fa2 = relu(facat[h] @ Wfc[h]^T + bfc[h]) -> faR[x][h*64+e]
// grid = (NX/16, NH), block = 32
// ---------------------------------------------------------------------------
__global__ void k_fc(const float* __restrict__ facat, const float* __restrict__ Wfc,
                     const float* __restrict__ bfc, float* __restrict__ faR) {
  const int h  = blockIdx.y;
  const int m0 = blockIdx.x * 16;

  v8f acc[4] = {};
  gemm_tile<4>(facat + (size_t)h * NX * 128, 128, m0,
               Wfc + (size_t)h * FATT * 128, 128, 0, 2 * FATT, acc);

  const int lane = threadIdx.x & 31;
  const int half = lane >> 4;
  const int r    = lane & 15;
#pragma unroll
  for (int s = 0; s < 4; ++s) {
    const int e = s * 16 + r;
    const float bb = bfc[h * FATT + e];
#pragma unroll
    for (int j = 0; j < 8; ++j) {
      const int x = m0 + j + 8 * half;
      faR[(size_t)x * FDIM + h * FATT + e] = fmaxf(acc[s][j] + bb, 0.f);
    }
  }
}

// ---------------------------------------------------------------------------
// Kernel 6: fx_out = relu(fx_in + faR @ Wout^T + bout)
// grid = (FDIM/64, NX/16), block = 32
// ---------------------------------------------------------------------------
__global__ void k_out(const float* __restrict__ faR, const float* __restrict__ Wout,
                      const float* __restrict__ bout, const float* __restrict__ fx_in,
                      float* __restrict__ out) {
  const int n0 = blockIdx.x * 64;
  const int m0 = blockIdx.y * 16;

  v8f acc[4] = {};
  gemm_tile<4>(faR, FDIM, m0, Wout, FDIM, n0, FDIM, acc);

  const int lane = threadIdx.x & 31;
  const int half = lane >> 4;
  const int r    = lane & 15;
#pragma unroll
  for (int s = 0; s < 4; ++s) {
    const int n = n0 + s * 16 + r;
    const float bb = bout[n];
#pragma unroll
    for (int j = 0; j < 8; ++j) {
      const int m = m0 + j + 8 * half;
      const float v = fx_in[(size_t)m * FDIM + n] + acc[s][j] + bb;
      out[(size_t)m * FDIM + n] = fmaxf(v, 0.f);
    }
  }
}

// ---------------------------------------------------------------------------
extern "C" void kernel_launch(void* const* d_in, const int* in_sizes, int n_in,
                              void* d_out, int out_size, void* d_ws, size_t ws_size,
                              hipStream_t stream) {
  (void)in_sizes; (void)n_in; (void)out_size; (void)ws_size;

  const float* fx_in = (const float*)d_in[0];
  const float* fp_in = (const float*)d_in[1];  // (1, NP, FDIM) -> fp0
  const float* Wembd = (const float*)d_in[2];  // (NH, FATT, FDIM)
  const float* bembd = (const float*)d_in[3];  // (NH, FATT)
  const float* Wfc   = (const float*)d_in[4];  // (NH, FATT, 2*FATT)
  const float* bfc   = (const float*)d_in[5];  // (NH, FATT)
  const float* Wout  = (const float*)d_in[6];  // (FDIM, FDIM)
  const float* bout  = (const float*)d_in[7];  // (FDIM,)

  float* out  = (float*)d_out;
  float* wout = out + (size_t)NX * FDIM;  // w_out region, layout (x, h, p)

  float* ws    = (float*)d_ws;
  float* fxE   = ws;                               // NH*NX*FATT
  float* fpE   = fxE  + (size_t)NH * NX * FATT;    // NH*NP*FATT
  float* fpT   = fpE  + (size_t)NH * NP * FATT;    // NH*FATT*NP
  float* facat = fpT  + (size_t)NH * FATT * NP;    // NH*NX*128
  // faR: NX*FDIM
  float* faR   = facat + (size_t)NH * NX * 128;

  k_embed<<<dim3(FDIM / 64, NTOT / 16), dim3(32), 0, stream>>>(
      fx_in, fp_in, Wembd, bembd, fxE, fpE, fpT);

  k_scores<<<dim3(NP / 64, NX / 16, NH), dim3(32), 0, stream>>>(fxE, fpE, wout);

  k_softmax<<<dim3(NX * NH), dim3(256), 0, stream>>>(wout);

  k_attv<<<dim3(NX / 16, NH), dim3(32), 0, stream>>>(wout, fpT, fxE, facat);

  k_fc<<<dim3(NX / 16, NH), dim3(32), 0, stream>>>(facat, Wfc, bfc, faR);

  k_out<<<dim3(FDIM / 64, NX / 16), dim3(32), 0, stream>>>(faR, Wout, bout, fx_in, out);
}